// GATModel_89541478187029
// MI455X (gfx1250) — compile-verified
//
#include <hip/hip_runtime.h>
#include <math.h>

#define BS   4
#define NN   15135
#define FIN  8
#define HD   64
#define NL   3
#define NE   302700
#define ET   (NE + NN)      // edges + self loops
#define FCD  256
#define NC   2
#define LREL 0.2f

typedef float v2f __attribute__((ext_vector_type(2)));
typedef float v8f __attribute__((ext_vector_type(8)));

__device__ __forceinline__ float leaky(float x) { return x > 0.f ? x : LREL * x; }

__device__ __forceinline__ float atomicMaxF(float* addr, float val) {
    int* ai = (int*)addr;
    int old = *ai;
    while (__int_as_float(old) < val) {
        int assumed = old;
        old = atomicCAS(ai, assumed, __float_as_int(val));
        if (old == assumed) break;
    }
    return __int_as_float(old);
}

__device__ __forceinline__ void edge_sd(const int* __restrict__ ei, int e, int& s, int& d) {
    if (e < NE) { s = ei[e]; d = ei[NE + e]; }
    else        { s = e - NE; d = e - NE; }     // self loop
}

// ---------------- layer 0: h = x @ W0  (Fin=8, trivially small K) ----------------
__global__ void k_layer0(const float* __restrict__ x, const float* __restrict__ W,
                         float* __restrict__ h) {
    long t = (long)blockIdx.x * blockDim.x + threadIdx.x;
    if (t >= (long)BS * NN * HD) return;
    int hc = (int)(t % HD);
    long bn = t / HD;
    const float* xr = x + bn * FIN;
    float s = 0.f;
#pragma unroll
    for (int f = 0; f < FIN; ++f) s += xr[f] * W[f * HD + hc];
    h[t] = s;
}

// ---------------- layers 1,2: h = feat[N x 64] @ W[64 x 64] via WMMA f32 16x16x4 ----
// One wave computes a 16-row x 64-col strip. 8 waves/block -> 128 rows/block.
__global__ __launch_bounds__(256) void k_gemm64(const float* __restrict__ A,
                                                const float* __restrict__ W,
                                                float* __restrict__ O) {
    __shared__ float Ws[HD * HD];
    int tid = threadIdx.x;
    for (int i = tid; i < HD * HD; i += 256) Ws[i] = W[i];
    __syncthreads();

    int wave = tid >> 5, lane = tid & 31;
    int b = blockIdx.y;
    long row0 = (long)blockIdx.x * 128 + wave * 16;
    const float* Ab = A + (long)b * NN * HD;
    float*       Ob = O + (long)b * NN * HD;

    int m  = lane & 15;       // row within 16-tile (A) / col within 16-tile (B,C)
    int kh = lane >> 4;       // K half select (0 -> k0,k0+1 ; 1 -> k0+2,k0+3)
    long arow = row0 + m;
    float sc = (arow < NN) ? 1.f : 0.f;           // zero out tail rows, keep EXEC uniform
    const float* ap = Ab + (arow < NN ? arow : 0) * HD;

    v8f acc[4] = {};
    for (int k0 = 0; k0 < HD; k0 += 4) {
        int kk = k0 + kh * 2;
        v2f a;
        a.x = ap[kk]     * sc;
        a.y = ap[kk + 1] * sc;
#pragma unroll
        for (int t4 = 0; t4 < 4; ++t4) {
            int c = t4 * 16 + m;
            v2f bv;
            bv.x = Ws[kk * HD + c];
            bv.y = Ws[(kk + 1) * HD + c];
            acc[t4] = __builtin_amdgcn_wmma_f32_16x16x4_f32(
                false, a, false, bv, (short)0, acc[t4], false, false);
        }
    }
    // C/D layout: VGPR r -> lanes 0-15: M=r, lanes 16-31: M=r+8; N = lane&15
#pragma unroll
    for (int t4 = 0; t4 < 4; ++t4)
#pragma unroll
        for (int r = 0; r < 8; ++r) {
            long row = row0 + r + kh * 8;
            if (row < NN) Ob[row * HD + t4 * 16 + m] = acc[t4][r];
        }
}

// ---------------- per-node attention scalars es = h.a_src, ed = h.a_dst ----------------
__global__ void k_attn(const float* __restrict__ h, const float* __restrict__ as,
                       const float* __restrict__ ad, float* __restrict__ es,
                       float* __restrict__ ed) {
    long t = (long)blockIdx.x * blockDim.x + threadIdx.x;
    if (t >= (long)BS * NN) return;
    const float* hr = h + t * HD;
    float s1 = 0.f, s2 = 0.f;
#pragma unroll 8
    for (int i = 0; i < HD; ++i) { float v = hr[i]; s1 += v * as[i]; s2 += v * ad[i]; }
    es[t] = s1; ed[t] = s2;
}

__global__ void k_init_nodes(float* __restrict__ smax, float* __restrict__ den) {
    long t = (long)blockIdx.x * blockDim.x + threadIdx.x;
    if (t >= (long)BS * NN) return;
    smax[t] = -INFINITY; den[t] = 0.f;
}

__global__ void k_zero(float* __restrict__ p, long n) {
    long t = (long)blockIdx.x * blockDim.x + threadIdx.x;
    if (t < n) p[t] = 0.f;
}

// ---------------- edge softmax pass 1: segment max over dst ----------------
__global__ void k_edge_max(const int* __restrict__ ei, const float* __restrict__ es,
                           const float* __restrict__ ed, float* __restrict__ smax) {
    long t = (long)blockIdx.x * blockDim.x + threadIdx.x;
    if (t >= (long)BS * ET) return;
    int b = (int)(t / ET), e = (int)(t % ET);
    int s, d; edge_sd(ei, e, s, d);
    float v = leaky(es[(long)b * NN + s] + ed[(long)b * NN + d]);
    atomicMaxF(&smax[(long)b * NN + d], v);
}

// ---------------- pass 2: ex = exp(e - max), segment sum denom ----------------
__global__ void k_edge_exp(const int* __restrict__ ei, const float* __restrict__ es,
                           const float* __restrict__ ed, const float* __restrict__ smax,
                           float* __restrict__ den, float* __restrict__ exb) {
    long t = (long)blockIdx.x * blockDim.x + threadIdx.x;
    if (t >= (long)BS * ET) return;
    int b = (int)(t / ET), e = (int)(t % ET);
    int s, d; edge_sd(ei, e, s, d);
    float v  = leaky(es[(long)b * NN + s] + ed[(long)b * NN + d]);
    float ex = __expf(v - smax[(long)b * NN + d]);
    exb[t] = ex;
    atomicAdd(&den[(long)b * NN + d], ex);
}

// ---------------- pass 3: acc[dst] += alpha * h[src]  (64 lanes per edge) ----------------
__global__ void k_edge_scatter(const int* __restrict__ ei, const float* __restrict__ exb,
                               const float* __restrict__ den, const float* __restrict__ h,
                               float* __restrict__ acc) {
    long t = (long)blockIdx.x * blockDim.x + threadIdx.x;
    if (t >= (long)BS * ET * HD) return;
    int hc = (int)(t & (HD - 1));
    long be = t >> 6;
    int b = (int)(be / ET), e = (int)(be % ET);
    int s, d; edge_sd(ei, e, s, d);
    float alpha = exb[be] / den[(long)b * NN + d];
    atomicAdd(&acc[((long)b * NN + d) * HD + hc],
              alpha * h[((long)b * NN + s) * HD + hc]);
}

// ---------------- relu(acc + bias), write next-layer feat (in place) and xcat ----------------
__global__ void k_finalize(float* __restrict__ acc, const float* __restrict__ bias,
                           float* __restrict__ xcat, int layer) {
    long t = (long)blockIdx.x * blockDim.x + threadIdx.x;
    if (t >= (long)BS * NN * HD) return;
    int hc = (int)(t % HD);
    long bn = t / HD;
    float v = acc[t] + bias[hc];
    v = v > 0.f ? v : 0.f;
    acc[t] = v;
    // stack(xs,-1).reshape -> feature index = h*NL + layer
    xcat[bn * (HD * NL) + hc * NL + layer] = v;
}

// ---------------- scores[b,n] = xcat[b,n,:] . fc_w + fc_b ----------------
__global__ void k_fc(const float* __restrict__ xcat, const float* __restrict__ fcw,
                     const float* __restrict__ fcb, float* __restrict__ scores) {
    long t = (long)blockIdx.x * blockDim.x + threadIdx.x;
    if (t >= (long)BS * NN) return;
    const float* xr = xcat + t * (HD * NL);
    float s = fcb[0];
#pragma unroll 8
    for (int i = 0; i < HD * NL; ++i) s += xr[i] * fcw[i];
    scores[t] = s;
}

// ---------------- h1 = relu(scores @ lin1_w + b) : [4,N] x [N,256] ----------------
__global__ void k_lin1(const float* __restrict__ scores, const float* __restrict__ w,
                       const float* __restrict__ bias, float* __restrict__ h1) {
    int t = blockIdx.x * blockDim.x + threadIdx.x;
    if (t >= BS * FCD) return;
    int f = t % FCD, b = t / FCD;
    const float* sr = scores + (long)b * NN;
    float s = bias[f];
    for (int n = 0; n < NN; ++n) {
        if ((n & 63) == 0) __builtin_prefetch(&w[(long)(n + 64) * FCD + f], 0, 1);
        s += sr[n] * w[(long)n * FCD + f];
    }
    h1[t] = fmaxf(s, 0.f);
}

// ---------------- logits = h1 @ lin2_w + b ; log_softmax ----------------
__global__ void k_head(const float* __restrict__ h1, const float* __restrict__ w2,
                       const float* __restrict__ b2, float* __restrict__ out) {
    __shared__ float lg[BS * NC];
    int t = threadIdx.x;
    if (t < BS * NC) {
        int c = t % NC, b = t / NC;
        const float* hr = h1 + b * FCD;
        float s = b2[c];
        for (int i = 0; i < FCD; ++i) s += hr[i] * w2[i * NC + c];
        lg[t] = s;
    }
    __syncthreads();
    if (t < BS * NC) {
        int b = t / NC;
        float l0 = lg[b * NC], l1 = lg[b * NC + 1];
        float m = fmaxf(l0, l1);
        float lse = m + logf(__expf(l0 - m) + __expf(l1 - m));
        out[t] = lg[t] - lse;
    }
}

extern "C" void kernel_launch(void* const* d_in, const int* in_sizes, int n_in,
                              void* d_out, int out_size, void* d_ws, size_t ws_size,
                              hipStream_t stream) {
    (void)in_sizes; (void)n_in; (void)out_size; (void)ws_size;
    const float* x   = (const float*)d_in[0];
    const int*   ei  = (const int*)d_in[1];
    const float* W[3]   = {(const float*)d_in[3],  (const float*)d_in[7],  (const float*)d_in[11]};
    const float* asv[3] = {(const float*)d_in[4],  (const float*)d_in[8],  (const float*)d_in[12]};
    const float* adv[3] = {(const float*)d_in[5],  (const float*)d_in[9],  (const float*)d_in[13]};
    const float* bv[3]  = {(const float*)d_in[6],  (const float*)d_in[10], (const float*)d_in[14]};
    const float* fcw = (const float*)d_in[15];
    const float* fcb = (const float*)d_in[16];
    const float* l1w = (const float*)d_in[17];
    const float* l1b = (const float*)d_in[18];
    const float* l2w = (const float*)d_in[19];
    const float* l2b = (const float*)d_in[20];

    const long BN  = (long)BS * NN;
    const long BNH = BN * HD;
    const long BE  = (long)BS * ET;

    float* ws     = (float*)d_ws;
    float* bufA   = ws;                 // B*N*64
    float* bufB   = bufA + BNH;         // B*N*64
    float* hb     = bufB + BNH;         // B*N*64
    float* es     = hb + BNH;           // B*N
    float* ed     = es + BN;            // B*N
    float* smax   = ed + BN;            // B*N
    float* den    = smax + BN;          // B*N
    float* exb    = den + BN;           // B*ET
    float* xcat   = exb + BE;           // B*N*192
    float* scores = xcat + BN * HD * NL;// B*N
    float* h1     = scores + BN;        // B*256

    dim3 blk(256);
    auto g1 = [](long n) { return dim3((unsigned)((n + 255) / 256)); };

    for (int l = 0; l < NL; ++l) {
        float* acc = (l == 1) ? bufB : bufA;            // l0->A, l1->B, l2->A
        const float* featin = (l == 1) ? bufA : bufB;   // l1 reads A, l2 reads B
        if (l == 0) {
            k_layer0<<<g1(BNH), blk, 0, stream>>>(x, W[0], hb);
        } else {
            dim3 grid((NN + 127) / 128, BS);
            k_gemm64<<<grid, 256, 0, stream>>>(featin, W[l], hb);
        }
        k_attn<<<g1(BN), blk, 0, stream>>>(hb, asv[l], adv[l], es, ed);
        k_init_nodes<<<g1(BN), blk, 0, stream>>>(smax, den);
        k_zero<<<g1(BNH), blk, 0, stream>>>(acc, BNH);
        k_edge_max<<<g1(BE), blk, 0, stream>>>(ei, es, ed, smax);
        k_edge_exp<<<g1(BE), blk, 0, stream>>>(ei, es, ed, smax, den, exb);
        k_edge_scatter<<<g1(BE * HD), blk, 0, stream>>>(ei, exb, den, hb, acc);
        k_finalize<<<g1(BNH), blk, 0, stream>>>(acc, bv[l], xcat, l);
    }
    k_fc<<<g1(BN), blk, 0, stream>>>(xcat, fcw, fcb, scores);
    k_lin1<<<g1(BS * FCD), blk, 0, stream>>>(scores, l1w, l1b, h1);
    k_head<<<1, 32, 0, stream>>>(h1, l2w, l2b, (float*)d_out);
}